// DecoderRNN_2241972929122
// MI455X (gfx1250) — compile-verified
//
#include <hip/hip_runtime.h>
#include <hip/hip_bf16.h>
#include <math.h>

typedef __attribute__((ext_vector_type(2))) float v2f;
typedef __attribute__((ext_vector_type(8))) float v8f;

#define B_   32
#define T_   64
#define H_   1024
#define V_   32000
#define G3H  3072
#define SOS_ 1

__device__ __forceinline__ v8f wmma_f32(v2f a, v2f b, v8f c) {
  // D = A(16x4 f32) * B(4x16 f32) + C(16x16 f32)
  return __builtin_amdgcn_wmma_f32_16x16x4_f32(
      /*neg_a=*/false, a, /*neg_b=*/false, b,
      /*c_mod=*/(short)0, c, /*reuse_a=*/false, /*reuse_b=*/false);
}

__device__ __forceinline__ float sigmoid_f(float x) {
  return 1.0f / (1.0f + __expf(-x));
}

// ---------------------------------------------------------------------------
// Kernel 1: GI = relu(embedding[tok]) @ W_ih^T + b_ih   for all (t,b) rows.
// Rows are (t*B + b), 2048 total. Gather+ReLU fused into A-fragment loads.
// Wave = 16x64 strip. Block = 8 waves = 128 rows x 64 cols.
// ---------------------------------------------------------------------------
__global__ void __launch_bounds__(256)
gi_gemm_kernel(const float* __restrict__ emb, const int* __restrict__ target,
               const float* __restrict__ W_ih, const float* __restrict__ b_ih,
               float* __restrict__ GI) {
  const int lane = threadIdx.x & 31;
  const int wave = threadIdx.x >> 5;
  const int m0   = (blockIdx.y * 8 + wave) * 16;
  const int n0   = blockIdx.x * 64;
  const int l15  = lane & 15;
  const int koff = (lane >> 4) << 1;

  // token for this lane's A row (teacher forcing: step 0 uses SOS)
  const int mrow = m0 + l15;
  const int t    = mrow >> 5;
  const int bidx = mrow & 31;
  const int tok  = (t == 0) ? SOS_ : target[bidx * T_ + (t - 1)];
  const float* __restrict__ Arow = emb + (size_t)tok * H_;
  const float* __restrict__ B0   = W_ih + (size_t)(n0 + l15) * H_;

  v8f acc0 = {}, acc1 = {}, acc2 = {}, acc3 = {};
  for (int k = 0; k < H_; k += 4) {
    const int kk = k + koff;
    v2f a = *(const v2f*)(Arow + kk);
    a.x = fmaxf(a.x, 0.0f);
    a.y = fmaxf(a.y, 0.0f);
    v2f b0 = *(const v2f*)(B0 + 0 * 16 * H_ + kk);
    v2f b1 = *(const v2f*)(B0 + 1 * 16 * H_ + kk);
    v2f b2 = *(const v2f*)(B0 + 2 * 16 * H_ + kk);
    v2f b3 = *(const v2f*)(B0 + 3 * 16 * H_ + kk);
    acc0 = wmma_f32(a, b0, acc0);
    acc1 = wmma_f32(a, b1, acc1);
    acc2 = wmma_f32(a, b2, acc2);
    acc3 = wmma_f32(a, b3, acc3);
  }

  const int rbase = m0 + ((lane >> 4) << 3);
#pragma unroll
  for (int r = 0; r < 8; ++r) {
    const int row = rbase + r;
    float* o = GI + (size_t)row * G3H + n0 + l15;
    o[0]  = acc0[r] + b_ih[n0 + 0  + l15];
    o[16] = acc1[r] + b_ih[n0 + 16 + l15];
    o[32] = acc2[r] + b_ih[n0 + 32 + l15];
    o[48] = acc3[r] + b_ih[n0 + 48 + l15];
  }
}

// ---------------------------------------------------------------------------
// Kernel 2 (x64, stream-serialized): one GRU step, fully fused.
// Each wave owns one 16x16 (m, j) tile and computes gh for r/z/n gates at
// that position, then the gate nonlinearity + h update in-register.
// Block = 8 waves = 32 rows x 64 j-cols.  Grid = H/64 = 16 blocks.
// ---------------------------------------------------------------------------
__global__ void __launch_bounds__(256)
gru_step_kernel(const float* __restrict__ Hprev, float* __restrict__ Hnext,
                const float* __restrict__ W_hh, const float* __restrict__ b_hh,
                const float* __restrict__ GI_t) {
  const int lane = threadIdx.x & 31;
  const int wave = threadIdx.x >> 5;
  const int m0   = (wave >> 2) * 16;            // 0 or 16
  const int j0   = blockIdx.x * 64 + (wave & 3) * 16;
  const int l15  = lane & 15;
  const int koff = (lane >> 4) << 1;

  const float* __restrict__ Arow = Hprev + (size_t)(m0 + l15) * H_;
  const int jc = j0 + l15;
  const float* __restrict__ Br = W_hh + (size_t)(0 * H_ + jc) * H_;
  const float* __restrict__ Bz = W_hh + (size_t)(1 * H_ + jc) * H_;
  const float* __restrict__ Bn = W_hh + (size_t)(2 * H_ + jc) * H_;

  v8f aR = {}, aZ = {}, aN = {};
  for (int k = 0; k < H_; k += 4) {
    const int kk = k + koff;
    v2f a  = *(const v2f*)(Arow + kk);
    v2f br = *(const v2f*)(Br + kk);
    v2f bz = *(const v2f*)(Bz + kk);
    v2f bn = *(const v2f*)(Bn + kk);
    aR = wmma_f32(a, br, aR);
    aZ = wmma_f32(a, bz, aZ);
    aN = wmma_f32(a, bn, aN);
  }

  const float bhr = b_hh[jc];
  const float bhz = b_hh[H_ + jc];
  const float bhn = b_hh[2 * H_ + jc];
  const int rbase = m0 + ((lane >> 4) << 3);
#pragma unroll
  for (int r = 0; r < 8; ++r) {
    const int row = rbase + r;                      // batch index 0..31
    const float* gi = GI_t + (size_t)row * G3H;
    const float rr = sigmoid_f(gi[jc]           + aR[r] + bhr);
    const float zz = sigmoid_f(gi[H_ + jc]      + aZ[r] + bhz);
    const float nn = tanhf(gi[2 * H_ + jc] + rr * (aN[r] + bhn));
    const float h  = Hprev[(size_t)row * H_ + jc];
    Hnext[(size_t)row * H_ + jc] = (1.0f - zz) * nn + zz * h;
  }
}

// ---------------------------------------------------------------------------
// Kernel 3: logits = H_all @ W_out^T + b_out  -- one batched [2048 x 32000]
// GEMM, K=1024.  Epilogue remaps GEMM row (t*B+b) -> output row (b*T+t).
// W_out (131 MB) fits in the 192 MB L2, read from HBM once.
// ---------------------------------------------------------------------------
__global__ void __launch_bounds__(256)
logits_gemm_kernel(const float* __restrict__ Hrows, const float* __restrict__ W_out,
                   const float* __restrict__ b_out, float* __restrict__ out) {
  const int lane = threadIdx.x & 31;
  const int wave = threadIdx.x >> 5;
  const int m0   = (blockIdx.y * 8 + wave) * 16;
  const int n0   = blockIdx.x * 64;
  const int l15  = lane & 15;
  const int koff = (lane >> 4) << 1;

  const float* __restrict__ Arow = Hrows + (size_t)(m0 + l15) * H_;
  const float* __restrict__ B0   = W_out + (size_t)(n0 + l15) * H_;

  v8f acc0 = {}, acc1 = {}, acc2 = {}, acc3 = {};
  for (int k = 0; k < H_; k += 4) {
    const int kk = k + koff;
    v2f a  = *(const v2f*)(Arow + kk);
    v2f b0 = *(const v2f*)(B0 + 0 * 16 * H_ + kk);
    v2f b1 = *(const v2f*)(B0 + 1 * 16 * H_ + kk);
    v2f b2 = *(const v2f*)(B0 + 2 * 16 * H_ + kk);
    v2f b3 = *(const v2f*)(B0 + 3 * 16 * H_ + kk);
    acc0 = wmma_f32(a, b0, acc0);
    acc1 = wmma_f32(a, b1, acc1);
    acc2 = wmma_f32(a, b2, acc2);
    acc3 = wmma_f32(a, b3, acc3);
  }

  const float bo0 = b_out[n0 + 0  + l15];
  const float bo1 = b_out[n0 + 16 + l15];
  const float bo2 = b_out[n0 + 32 + l15];
  const float bo3 = b_out[n0 + 48 + l15];
  const int rbase = m0 + ((lane >> 4) << 3);
#pragma unroll
  for (int r = 0; r < 8; ++r) {
    const int row = rbase + r;           // = t*B + b
    const int t   = row >> 5;
    const int b   = row & 31;
    float* o = out + (size_t)(b * T_ + t) * V_ + n0 + l15;
    o[0]  = acc0[r] + bo0;
    o[16] = acc1[r] + bo1;
    o[32] = acc2[r] + bo2;
    o[48] = acc3[r] + bo3;
  }
}

// ---------------------------------------------------------------------------
// Kernel 4: in-place log_softmax over V for each of the 2048 output rows.
// ---------------------------------------------------------------------------
__global__ void __launch_bounds__(256)
log_softmax_kernel(float* __restrict__ out) {
  float* p = out + (size_t)blockIdx.x * V_;
  float4* p4 = (float4*)p;
  const int tid = threadIdx.x;
  __shared__ float sred[256];

  float mx = -3.402823466e38f;
  for (int i = tid; i < V_ / 4; i += 256) {
    float4 v = p4[i];
    mx = fmaxf(mx, fmaxf(fmaxf(v.x, v.y), fmaxf(v.z, v.w)));
  }
  sred[tid] = mx;
  __syncthreads();
  for (int s = 128; s > 0; s >>= 1) {
    if (tid < s) sred[tid] = fmaxf(sred[tid], sred[tid + s]);
    __syncthreads();
  }
  mx = sred[0];
  __syncthreads();

  float sum = 0.0f;
  for (int i = tid; i < V_ / 4; i += 256) {
    float4 v = p4[i];
    sum += __expf(v.x - mx) + __expf(v.y - mx) + __expf(v.z - mx) + __expf(v.w - mx);
  }
  sred[tid] = sum;
  __syncthreads();
  for (int s = 128; s > 0; s >>= 1) {
    if (tid < s) sred[tid] += sred[tid + s];
    __syncthreads();
  }
  const float lse = mx + __logf(sred[0]);
  for (int i = tid; i < V_ / 4; i += 256) {
    float4 v = p4[i];
    v.x -= lse; v.y -= lse; v.z -= lse; v.w -= lse;
    p4[i] = v;
  }
}

// ---------------------------------------------------------------------------
// Launch. Workspace layout (floats):
//   GI   : [T*B, 3H]       = 6,291,456 floats (~25.2 MB)
//   Hseq : [(T+1), B, H]   = 2,129,920 floats (~8.5 MB)   (Hseq[0] = h0)
// ---------------------------------------------------------------------------
extern "C" void kernel_launch(void* const* d_in, const int* in_sizes, int n_in,
                              void* d_out, int out_size, void* d_ws, size_t ws_size,
                              hipStream_t stream) {
  (void)in_sizes; (void)n_in; (void)out_size; (void)ws_size;

  const float* enc_hidden = (const float*)d_in[1];   // [1, B, H]
  const int*   target     = (const int*)d_in[2];     // [B, T]
  const float* emb        = (const float*)d_in[4];   // [V, H]
  const float* W_ih       = (const float*)d_in[5];   // [3H, H]
  const float* W_hh       = (const float*)d_in[6];   // [3H, H]
  const float* b_ih       = (const float*)d_in[7];   // [3H]
  const float* b_hh       = (const float*)d_in[8];   // [3H]
  const float* W_out      = (const float*)d_in[9];   // [V, H]
  const float* b_out      = (const float*)d_in[10];  // [V]
  float* out = (float*)d_out;                        // [B,T,V] ++ [1,B,H]

  float* GI   = (float*)d_ws;                        // [2048, 3072]
  float* Hseq = GI + (size_t)T_ * B_ * G3H;          // [(T+1), B, H]

  // Hseq[0] = encoder_hidden[0]
  hipMemcpyAsync(Hseq, enc_hidden, (size_t)B_ * H_ * sizeof(float),
                 hipMemcpyDeviceToDevice, stream);

  const dim3 blk(256);

  // 1) all-steps input-gate GEMM (gather + relu fused)
  gi_gemm_kernel<<<dim3(G3H / 64, (T_ * B_) / 128), blk, 0, stream>>>(
      emb, target, W_ih, b_ih, GI);

  // 2) sequential recurrence (stream-ordered)
  for (int t = 0; t < T_; ++t) {
    gru_step_kernel<<<dim3(H_ / 64), blk, 0, stream>>>(
        Hseq + (size_t)t * B_ * H_, Hseq + (size_t)(t + 1) * B_ * H_,
        W_hh, b_hh, GI + (size_t)t * B_ * G3H);
  }

  // 3) one batched output projection over all 2048 (t,b) rows
  logits_gemm_kernel<<<dim3(V_ / 64, (T_ * B_) / 128), blk, 0, stream>>>(
      Hseq + (size_t)B_ * H_, W_out, b_out, out);

  // 4) in-place log_softmax
  log_softmax_kernel<<<dim3(B_ * T_), blk, 0, stream>>>(out);

  // 5) h_final = Hseq[T]
  hipMemcpyAsync(out + (size_t)B_ * T_ * V_, Hseq + (size_t)T_ * B_ * H_,
                 (size_t)B_ * H_ * sizeof(float), hipMemcpyDeviceToDevice, stream);
}